// PointNetPPInstSeg_90185723281837
// MI455X (gfx1250) — compile-verified
//
#include <hip/hip_runtime.h>
#include <hip/hip_bf16.h>
#include <stdint.h>

// ---------------------------------------------------------------------------
// Problem constants (from the reference setup_inputs): bz=32, N=16384,
// nsmp=4096, K=64, C=6.  Output: [bz, nsmp, 11] fp32.
// ---------------------------------------------------------------------------
constexpr int K_NB  = 64;
constexpr int C_F   = 6;
constexpr int NSMP  = 4096;      // samples per batch (power of two -> shift)
constexpr int NPTS  = 16384;     // points per batch
constexpr int BZ    = 32;
constexpr int SPB   = 8;         // samples per block = 1 per wave (8 waves/block)
constexpr int OUTC  = 11;

// ---------------------------------------------------------------------------
// Tensor Data Mover staging (gfx1250).  1-D copy of `nelem` 4-byte elements
// from global memory to LDS, descriptor packed per CDNA5 ISA ch. 8.
// This toolchain declares the 6-arg builtin:
//   (uint32x4 g0, int32x8 g1, int32x4 g2, int32x4 g3, int32x8 pad, i32 cpol)
// ---------------------------------------------------------------------------
typedef unsigned int u32x4 __attribute__((ext_vector_type(4)));
typedef int          i32x8 __attribute__((ext_vector_type(8)));
typedef int          i32x4 __attribute__((ext_vector_type(4)));

#ifndef __has_builtin
#define __has_builtin(x) 0
#endif
#if __has_builtin(__builtin_amdgcn_tensor_load_to_lds) && \
    __has_builtin(__builtin_amdgcn_s_wait_tensorcnt)
#define USE_TDM 1
#else
#define USE_TDM 0
#endif

#if USE_TDM
__device__ __forceinline__ void tdm_load_1d_b32(void* lds_dst, const void* gsrc,
                                                unsigned nelem) {
  const unsigned lds_addr = (unsigned)(uintptr_t)lds_dst;   // low 32b = LDS offset
  const uint64_t ga       = (uint64_t)(uintptr_t)gsrc;      // 57-bit global addr

  u32x4 g0;
  g0[0] = 1u;                                         // count=1, user desc
  g0[1] = lds_addr;                                   // lds_addr[31:0]
  g0[2] = (unsigned)(ga & 0xFFFFFFFFull);             // global_addr[31:0]
  g0[3] = (unsigned)((ga >> 32) & 0x01FFFFFFull)      // global_addr[56:32]
        | 0x80000000u;                                // type=2 ("image")

  i32x8 g1;
  g1[0] = (int)(2u << 16);                            // data_size=2 -> 4 bytes
  g1[1] = (int)((nelem & 0xFFFFu) << 16);             // tensor_dim0[15:0] @ bit48
  g1[2] = (int)(((nelem >> 16) & 0xFFFFu)             // tensor_dim0[31:16]
        | (1u << 16));                                // tensor_dim1 = 1
  g1[3] = (int)((nelem & 0xFFFFu) << 16);             // tile_dim0 @ bit112
  g1[4] = 0;                                          // tile_dim1=0, tile_dim2=0
  g1[5] = (int)nelem;                                 // tensor_dim0_stride[31:0]
  g1[6] = 0;                                          // stride0 hi / stride1 lo
  g1[7] = 0;                                          // tensor_dim1_stride hi

  const i32x4 z4 = {0, 0, 0, 0};                      // groups 2/3 unused (<=2D)
  const i32x8 z8 = {0, 0, 0, 0, 0, 0, 0, 0};
  __builtin_amdgcn_tensor_load_to_lds(g0, g1, z4, z4, z8, 0);
}
#endif

__device__ __forceinline__ float wave_sum(float v) {
#pragma unroll
  for (int off = 16; off > 0; off >>= 1) v += __shfl_xor(v, off, 32);
  return v;
}

// ---------------------------------------------------------------------------
// One wave32 per sample.  Each lane processes 2 of the 64 neighbors.
// ---------------------------------------------------------------------------
__global__ __launch_bounds__(256, 4)
void PointNetPPInstSeg_frag_kernel(const float* __restrict__ features,
                                   const int*   __restrict__ nk_idx,
                                   const float* __restrict__ nk_dist,
                                   const int*   __restrict__ fps_idx,
                                   float*       __restrict__ out) {
  __shared__ int   s_idx[SPB * K_NB];    // 2 KB
  __shared__ float s_dist[SPB * K_NB];   // 2 KB

  const int tid  = threadIdx.x;
  const int lane = tid & 31;
  const int wave = tid >> 5;
  const long long gs0 = (long long)blockIdx.x * SPB;   // first sample of block

#if USE_TDM
  if (wave == 0) {                       // TDM ignores EXEC; issue from one wave
    tdm_load_1d_b32(s_idx,  nk_idx  + gs0 * K_NB, SPB * K_NB);
    tdm_load_1d_b32(s_dist, nk_dist + gs0 * K_NB, SPB * K_NB);
    __builtin_amdgcn_s_wait_tensorcnt(0);
  }
#else
  for (int i = tid; i < SPB * K_NB; i += 256) {
    s_idx[i]  = nk_idx[gs0 * K_NB + i];
    s_dist[i] = nk_dist[gs0 * K_NB + i];
  }
#endif
  __syncthreads();

  const long long gs = gs0 + wave;                 // global sample id
  const int b = (int)(gs >> 12);                   // gs / NSMP (nsmp = 4096)

  // feats_sub: 6 floats at row fps_idx[gs] of flattened [bz*N, C] features.
  const int frow = fps_idx[gs];                    // wave-uniform
  const float2* fsp =
      reinterpret_cast<const float2*>(features + (long long)frow * C_F);
  const float2 f01 = fsp[0], f23 = fsp[1], f45 = fsp[2];
  const float fs0 = f01.x, fs1 = f01.y, fs2 = f23.x;
  const float fs3 = f23.y, fs4 = f45.x, fs5 = f45.y;
  const float n_xyz = sqrtf(fs0 * fs0 + fs1 * fs1 + fs2 * fs2);
  const float n_nrm = sqrtf(fs3 * fs3 + fs4 * fs4 + fs5 * fs5);

  const float* fb = features + (long long)b * (NPTS * C_F);

  float cnt = 0.f, m0 = 0.f, m1 = 0.f, m2 = 0.f, m3 = 0.f, m4 = 0.f, m5 = 0.f;
  float sA = 0.f, sB = 0.f, cx = 0.f, cy = 0.f, cz = 0.f;

#pragma unroll
  for (int t = 0; t < K_NB / 32; ++t) {
    const int li = wave * K_NB + lane + 32 * t;
    const int nb = s_idx[li];
    const float d = s_dist[li];

    // gather neighbor row: 24 B, 8-byte aligned -> 3x global_load_b64 (L2 hit)
    const float2* np = reinterpret_cast<const float2*>(fb + (long long)nb * C_F);
    const float2 a = np[0], bq = np[1], c2 = np[2];
    const float v0 = a.x, v1 = a.y, v2 = bq.x, v3 = bq.y, v4 = c2.x, v5 = c2.y;

    // discrepancy: masked sums, r = 0.05
    const float ind = (d <= 0.05f) ? 1.0f : 0.0f;
    cnt += ind;
    m0 += v0 * ind; m1 += v1 * ind; m2 += v2 * ind;
    m3 += v3 * ind; m4 += v4 * ind; m5 += v5 * ind;

    // cosine similarity (per-neighbor ratio, then mean)
    const float num3 = fs0 * v0 + fs1 * v1 + fs2 * v2;
    const float den3 = fmaxf(n_xyz * sqrtf(v0 * v0 + v1 * v1 + v2 * v2), 1e-9f);
    sA += num3 / den3;
    const float numN = fs3 * v3 + fs4 * v4 + fs5 * v5;
    const float denN = fmaxf(n_nrm * sqrtf(v3 * v3 + v4 * v4 + v5 * v5), 1e-9f);
    sB += numN / denN;

    // cross(feats_sub.xyz, nk.xyz)
    cx += fs1 * v2 - fs2 * v1;
    cy += fs2 * v0 - fs0 * v2;
    cz += fs0 * v1 - fs1 * v0;
  }

  cnt = wave_sum(cnt);
  m0 = wave_sum(m0); m1 = wave_sum(m1); m2 = wave_sum(m2);
  m3 = wave_sum(m3); m4 = wave_sum(m4); m5 = wave_sum(m5);
  sA = wave_sum(sA); sB = wave_sum(sB);
  cx = wave_sum(cx); cy = wave_sum(cy); cz = wave_sum(cz);

  if (lane == 0) {
    float* o = out + gs * OUTC;
    const float invc = 1.0f / cnt;          // reference divides directly
    const float invk = 1.0f / (float)K_NB;
    o[0] = fs0 - m0 * invc;
    o[1] = fs1 - m1 * invc;
    o[2] = fs2 - m2 * invc;
    o[3] = fs3 - m3 * invc;
    o[4] = fs4 - m4 * invc;
    o[5] = fs5 - m5 * invc;
    o[6] = sA * invk;
    o[7] = sB * invk;
    o[8] = cx * invk;
    o[9] = cy * invk;
    o[10] = cz * invk;
  }
}

// ---------------------------------------------------------------------------
extern "C" void kernel_launch(void* const* d_in, const int* in_sizes, int n_in,
                              void* d_out, int out_size, void* d_ws,
                              size_t ws_size, hipStream_t stream) {
  const float* features = (const float*)d_in[0];
  const int*   nk_idx   = (const int*)d_in[1];
  const float* nk_dist  = (const float*)d_in[2];
  const int*   fps_idx  = (const int*)d_in[3];
  float*       out      = (float*)d_out;

  const int total_samples = BZ * NSMP;               // 131072
  const int blocks = total_samples / SPB;            // 16384
  PointNetPPInstSeg_frag_kernel<<<blocks, 256, 0, stream>>>(
      features, nk_idx, nk_dist, fps_idx, out);
}